// PreTrainedBertModel_65231963292404
// MI455X (gfx1250) — compile-verified
//
#include <hip/hip_runtime.h>
#include <hip/hip_bf16.h>
#include <math.h>

typedef __attribute__((ext_vector_type(2))) float v2f;
typedef __attribute__((ext_vector_type(8))) float v8f;

constexpr int kS   = 1024;
constexpr int kHID = 1024;
constexpr int kNH  = 16;
constexpr int kDH  = 64;
constexpr int kNS  = 2;
constexpr int kP   = 128;
constexpr int kRO  = 32;
constexpr int kMK  = kS + kP;   // 1152

__device__ __forceinline__ v8f wmma_f32(v2f a, v2f b, v8f c) {
  // V_WMMA_F32_16X16X4_F32 : D = A(16x4) * B(4x16) + C(16x16)
  return __builtin_amdgcn_wmma_f32_16x16x4_f32(false, a, false, b, (short)0, c,
                                               false, false);
}

__device__ __forceinline__ v8f vzero8() {
  v8f z = {0.f, 0.f, 0.f, 0.f, 0.f, 0.f, 0.f, 0.f};
  return z;
}

// ---------------------------------------------------------------------------
// QKV projection: out[r, j] = sum_d X[r,d] * W_qkv[j,d]  (j in [0,3072))
// Q = cols [0,1024)+q_bias, K = [1024,2048), V = [2048,3072)+v_bias
// grid = (R/16, 3072/128), block = 256 (8 waves); wave w -> 16x16 tile.
// ---------------------------------------------------------------------------
__global__ __launch_bounds__(256) void qkv_kernel(
    const float* __restrict__ X, const float* __restrict__ W,
    const float* __restrict__ q_bias, const float* __restrict__ v_bias,
    float* __restrict__ Q, float* __restrict__ K, float* __restrict__ V) {
  const int tid  = threadIdx.x;
  const int w    = tid >> 5;
  const int lane = tid & 31;
  const int half = lane >> 4;
  const int m    = lane & 15;

  const int row0 = blockIdx.x * 16;
  const int col0 = blockIdx.y * 128 + w * 16;

  const float* xp = X + (size_t)(row0 + m) * kHID + half * 2;
  const float* wp = W + (size_t)(col0 + m) * kHID + half * 2;

  v8f acc = vzero8();
  #pragma unroll 8
  for (int kk = 0; kk < kHID / 4; ++kk) {
    v2f a, b;
    a[0] = xp[kk * 4];  a[1] = xp[kk * 4 + 1];
    b[0] = wp[kk * 4];  b[1] = wp[kk * 4 + 1];
    acc = wmma_f32(a, b, acc);
  }

  #pragma unroll
  for (int v = 0; v < 8; ++v) {
    const int r = row0 + v + half * 8;
    const int c = col0 + m;
    const float val = acc[v];
    if (c < kHID) {
      Q[(size_t)r * kHID + c] = val + q_bias[c];
    } else if (c < 2 * kHID) {
      K[(size_t)r * kHID + (c - kHID)] = val;
    } else {
      V[(size_t)r * kHID + (c - 2 * kHID)] = val + v_bias[c - 2 * kHID];
    }
  }
}

// ---------------------------------------------------------------------------
// unary[h, k] = sum_d hidden_states[k, d] * W_unary[h, d]
// one wave per (h,k); grid = 16*1024/8 blocks of 256.
// ---------------------------------------------------------------------------
__global__ __launch_bounds__(256) void unary_kernel(
    const float* __restrict__ hs, const float* __restrict__ W_unary,
    float* __restrict__ unary) {
  const int wid  = blockIdx.x * 8 + (threadIdx.x >> 5);
  const int lane = threadIdx.x & 31;
  const int h = wid >> 10;
  const int k = wid & 1023;
  const float* xp = hs + (size_t)k * kHID;
  const float* wp = W_unary + (size_t)h * kHID;
  float s = 0.f;
  for (int i = lane; i < kHID; i += 32) s += xp[i] * wp[i];
  s += __shfl_xor(s, 1, 32);
  s += __shfl_xor(s, 2, 32);
  s += __shfl_xor(s, 4, 32);
  s += __shfl_xor(s, 8, 32);
  s += __shfl_xor(s, 16, 32);
  if (lane == 0) unary[h * kS + k] = s;
}

// ---------------------------------------------------------------------------
// Flash attention, main path.  One WG per (query-block of 16, head).
// Each wave handles column tiles ct = w, w+8, ... with online softmax;
// ctx accumulator (16x64) lives in 4 v8f per lane; 8-wave combine via LDS.
// ---------------------------------------------------------------------------
__global__ __launch_bounds__(256) void attn_main_kernel(
    const float* __restrict__ Qm, const float* __restrict__ Km,
    const float* __restrict__ Vm, const float* __restrict__ rel_pos,
    const float* __restrict__ W_rel, const float* __restrict__ unary,
    const unsigned char* __restrict__ mask, float* __restrict__ out) {
  __shared__ float Qs[16][64];
  __shared__ float pT[8][16][18];
  __shared__ float wrel[kRO];
  __shared__ float mL[8][16];
  __shared__ float lL[8][16];
  __shared__ float cL[8][16][64];

  const int qb = blockIdx.x, h = blockIdx.y;
  const int tid  = threadIdx.x;
  const int w    = tid >> 5;
  const int lane = tid & 31;
  const int half = lane >> 4;
  const int m    = lane & 15;
  const float scale = 0.125f;  // 1/sqrt(64)

  for (int i = tid; i < 16 * 64; i += 256) {
    const int r = i >> 6, c = i & 63;
    Qs[r][c] = Qm[(size_t)(qb * 16 + r) * kHID + h * kDH + c];
  }
  if (tid < kRO) wrel[tid] = W_rel[h * kRO + tid];
  __syncthreads();

  float mr[8], lr[8];
  v8f ctx[4];
  #pragma unroll
  for (int v = 0; v < 8; ++v) { mr[v] = -3.0e38f; lr[v] = 0.f; }
  #pragma unroll
  for (int t = 0; t < 4; ++t) ctx[t] = vzero8();

  const float* qrow = &Qs[m][half * 2];

  for (int ct = w; ct < kS / 16; ct += 8) {
    const int colbase = ct * 16;
    const int kcol = colbase + m;

    // --- scores: S = Q(16x64) * K^T(64x16) ---
    v8f sacc = vzero8();
    const float* kp = Km + (size_t)kcol * kHID + h * kDH + half * 2;
    #pragma unroll
    for (int kk = 0; kk < kDH / 4; ++kk) {
      v2f a, b;
      a[0] = qrow[kk * 4];  a[1] = qrow[kk * 4 + 1];
      b[0] = kp[kk * 4];    b[1] = kp[kk * 4 + 1];
      sacc = wmma_f32(a, b, sacc);
    }

    // --- epilogue: scale + rel-pos bias + key-unary + mask ---
    float p[8];
    const float un = unary[h * kS + kcol];
    #pragma unroll
    for (int v = 0; v < 8; ++v) {
      const int q = qb * 16 + v + half * 8;
      float sc = sacc[v] * scale;
      const float* rp = rel_pos + ((size_t)q * kS + kcol) * kRO;
      float rsum = 0.f;
      #pragma unroll
      for (int o = 0; o < kRO; ++o) rsum += rp[o] * wrel[o];
      sc += rsum + un;
      if (mask[(size_t)q * kS + kcol]) sc = -3.0e38f;
      p[v] = sc;
    }

    // --- online softmax update (rows split across the two 16-lane halves) ---
    #pragma unroll
    for (int v = 0; v < 8; ++v) {
      float rowmax = p[v];
      rowmax = fmaxf(rowmax, __shfl_xor(rowmax, 1, 32));
      rowmax = fmaxf(rowmax, __shfl_xor(rowmax, 2, 32));
      rowmax = fmaxf(rowmax, __shfl_xor(rowmax, 4, 32));
      rowmax = fmaxf(rowmax, __shfl_xor(rowmax, 8, 32));
      const float mnew = fmaxf(mr[v], rowmax);
      const float rsc  = __expf(mr[v] - mnew);
      const float pe   = __expf(p[v] - mnew);
      float rowsum = pe;
      rowsum += __shfl_xor(rowsum, 1, 32);
      rowsum += __shfl_xor(rowsum, 2, 32);
      rowsum += __shfl_xor(rowsum, 4, 32);
      rowsum += __shfl_xor(rowsum, 8, 32);
      lr[v] = lr[v] * rsc + rowsum;
      mr[v] = mnew;
      p[v]  = pe;
      #pragma unroll
      for (int t = 0; t < 4; ++t) ctx[t][v] *= rsc;
    }

    // --- bounce P tile through LDS (D-layout -> A-layout) ---
    #pragma unroll
    for (int v = 0; v < 8; ++v) pT[w][v + half * 8][m] = p[v];
    asm volatile("s_wait_dscnt 0" ::: "memory");

    // --- ctx += P(16x16) * V(16x64) ---
    #pragma unroll
    for (int t = 0; t < 4; ++t) {
      #pragma unroll
      for (int kk = 0; kk < 4; ++kk) {
        v2f a, b;
        a[0] = pT[w][m][kk * 4 + half * 2];
        a[1] = pT[w][m][kk * 4 + half * 2 + 1];
        const float* vp = Vm + (size_t)(colbase + kk * 4 + half * 2) * kHID +
                          h * kDH + t * 16 + m;
        b[0] = vp[0];
        b[1] = vp[kHID];
        ctx[t] = wmma_f32(a, b, ctx[t]);
      }
    }
  }

  // --- 8-wave combine via LDS ---
  if (m == 0) {
    #pragma unroll
    for (int v = 0; v < 8; ++v) {
      mL[w][v + half * 8] = mr[v];
      lL[w][v + half * 8] = lr[v];
    }
  }
  #pragma unroll
  for (int t = 0; t < 4; ++t)
    #pragma unroll
    for (int v = 0; v < 8; ++v) cL[w][v + half * 8][t * 16 + m] = ctx[t][v];
  __syncthreads();

  for (int e = tid; e < 16 * 64; e += 256) {
    const int r = e >> 6, d = e & 63;
    float M = -3.0e38f;
    #pragma unroll
    for (int ww = 0; ww < 8; ++ww) M = fmaxf(M, mL[ww][r]);
    float L = 0.f, val = 0.f;
    #pragma unroll
    for (int ww = 0; ww < 8; ++ww) {
      const float f = __expf(mL[ww][r] - M);
      L   += lL[ww][r] * f;
      val += cL[ww][r][d] * f;
    }
    out[(size_t)(qb * 16 + r) * kHID + h * kDH + d] = val / L;
  }
}

// ---------------------------------------------------------------------------
// Flash attention, two-stream predict path over merged keys (1024 main + 128
// predict).  One WG per (pblock of 16, head, ns).  No unary term (faithful).
// ---------------------------------------------------------------------------
__global__ __launch_bounds__(256) void attn_pred_kernel(
    const float* __restrict__ PQ, const float* __restrict__ PK,
    const float* __restrict__ PV, const float* __restrict__ Km,
    const float* __restrict__ Vm, const float* __restrict__ prel,
    const float* __restrict__ W_rel, const unsigned char* __restrict__ pmask,
    float* __restrict__ out) {
  __shared__ float Qs[16][64];
  __shared__ float pT[8][16][18];
  __shared__ float wrel[kRO];
  __shared__ float mL[8][16];
  __shared__ float lL[8][16];
  __shared__ float cL[8][16][64];

  const int pb = blockIdx.x, h = blockIdx.y, ns = blockIdx.z;
  const int tid  = threadIdx.x;
  const int w    = tid >> 5;
  const int lane = tid & 31;
  const int half = lane >> 4;
  const int m    = lane & 15;
  const float scale = 0.125f;

  for (int i = tid; i < 16 * 64; i += 256) {
    const int r = i >> 6, c = i & 63;
    Qs[r][c] = PQ[(size_t)(ns * kP + pb * 16 + r) * kHID + h * kDH + c];
  }
  if (tid < kRO) wrel[tid] = W_rel[h * kRO + tid];
  __syncthreads();

  float mr[8], lr[8];
  v8f ctx[4];
  #pragma unroll
  for (int v = 0; v < 8; ++v) { mr[v] = -3.0e38f; lr[v] = 0.f; }
  #pragma unroll
  for (int t = 0; t < 4; ++t) ctx[t] = vzero8();

  const float* qrow = &Qs[m][half * 2];

  for (int ct = w; ct < kMK / 16; ct += 8) {
    const int colbase = ct * 16;
    const int kcol = colbase + m;
    // tile is entirely main-keys or entirely predict-keys (boundary % 16 == 0)
    const float* kbase = (colbase < kS)
        ? (Km + (size_t)colbase * kHID)
        : (PK + (size_t)(ns * kP + colbase - kS) * kHID);
    const float* vbase = (colbase < kS)
        ? (Vm + (size_t)colbase * kHID)
        : (PV + (size_t)(ns * kP + colbase - kS) * kHID);

    // --- scores ---
    v8f sacc = vzero8();
    const float* kp = kbase + (size_t)m * kHID + h * kDH + half * 2;
    #pragma unroll
    for (int kk = 0; kk < kDH / 4; ++kk) {
      v2f a, b;
      a[0] = qrow[kk * 4];  a[1] = qrow[kk * 4 + 1];
      b[0] = kp[kk * 4];    b[1] = kp[kk * 4 + 1];
      sacc = wmma_f32(a, b, sacc);
    }

    // --- epilogue: scale + predict rel-pos bias + mask ---
    float p[8];
    #pragma unroll
    for (int v = 0; v < 8; ++v) {
      const int prow = pb * 16 + v + half * 8;
      float sc = sacc[v] * scale;
      const float* rp =
          prel + ((size_t)(ns * kP + prow) * kMK + kcol) * kRO;
      float rsum = 0.f;
      #pragma unroll
      for (int o = 0; o < kRO; ++o) rsum += rp[o] * wrel[o];
      sc += rsum;
      if (pmask[(size_t)(ns * kP + prow) * kMK + kcol]) sc = -3.0e38f;
      p[v] = sc;
    }

    // --- online softmax update ---
    #pragma unroll
    for (int v = 0; v < 8; ++v) {
      float rowmax = p[v];
      rowmax = fmaxf(rowmax, __shfl_xor(rowmax, 1, 32));
      rowmax = fmaxf(rowmax, __shfl_xor(rowmax, 2, 32));
      rowmax = fmaxf(rowmax, __shfl_xor(rowmax, 4, 32));
      rowmax = fmaxf(rowmax, __shfl_xor(rowmax, 8, 32));
      const float mnew = fmaxf(mr[v], rowmax);
      const float rsc  = __expf(mr[v] - mnew);
      const float pe   = __expf(p[v] - mnew);
      float rowsum = pe;
      rowsum += __shfl_xor(rowsum, 1, 32);
      rowsum += __shfl_xor(rowsum, 2, 32);
      rowsum += __shfl_xor(rowsum, 4, 32);
      rowsum += __shfl_xor(rowsum, 8, 32);
      lr[v] = lr[v] * rsc + rowsum;
      mr[v] = mnew;
      p[v]  = pe;
      #pragma unroll
      for (int t = 0; t < 4; ++t) ctx[t][v] *= rsc;
    }

    #pragma unroll
    for (int v = 0; v < 8; ++v) pT[w][v + half * 8][m] = p[v];
    asm volatile("s_wait_dscnt 0" ::: "memory");

    // --- ctx += P * V(merged) ---
    #pragma unroll
    for (int t = 0; t < 4; ++t) {
      #pragma unroll
      for (int kk = 0; kk < 4; ++kk) {
        v2f a, b;
        a[0] = pT[w][m][kk * 4 + half * 2];
        a[1] = pT[w][m][kk * 4 + half * 2 + 1];
        const float* vp = vbase + (size_t)(kk * 4 + half * 2) * kHID +
                          h * kDH + t * 16 + m;
        b[0] = vp[0];
        b[1] = vp[kHID];
        ctx[t] = wmma_f32(a, b, ctx[t]);
      }
    }
  }

  if (m == 0) {
    #pragma unroll
    for (int v = 0; v < 8; ++v) {
      mL[w][v + half * 8] = mr[v];
      lL[w][v + half * 8] = lr[v];
    }
  }
  #pragma unroll
  for (int t = 0; t < 4; ++t)
    #pragma unroll
    for (int v = 0; v < 8; ++v) cL[w][v + half * 8][t * 16 + m] = ctx[t][v];
  __syncthreads();

  for (int e = tid; e < 16 * 64; e += 256) {
    const int r = e >> 6, d = e & 63;
    float M = -3.0e38f;
    #pragma unroll
    for (int ww = 0; ww < 8; ++ww) M = fmaxf(M, mL[ww][r]);
    float L = 0.f, val = 0.f;
    #pragma unroll
    for (int ww = 0; ww < 8; ++ww) {
      const float f = __expf(mL[ww][r] - M);
      L   += lL[ww][r] * f;
      val += cL[ww][r][d] * f;
    }
    out[(size_t)(ns * kP + pb * 16 + r) * kHID + h * kDH + d] = val / L;
  }
}

// ---------------------------------------------------------------------------
extern "C" void kernel_launch(void* const* d_in, const int* in_sizes, int n_in,
                              void* d_out, int out_size, void* d_ws,
                              size_t ws_size, hipStream_t stream) {
  const float* hs            = (const float*)d_in[0];
  const unsigned char* amask = (const unsigned char*)d_in[1];
  const float* pseq          = (const float*)d_in[2];
  const unsigned char* pmask = (const unsigned char*)d_in[3];
  const float* rel           = (const float*)d_in[4];
  const float* prel          = (const float*)d_in[5];
  const float* Wqkv          = (const float*)d_in[6];
  const float* qb            = (const float*)d_in[7];
  const float* vb            = (const float*)d_in[8];
  const float* Wrel          = (const float*)d_in[9];
  const float* Wun           = (const float*)d_in[10];
  float* out = (float*)d_out;

  float* ws = (float*)d_ws;
  float* Qm = ws;                       // 1024*1024
  float* Km = Qm + (size_t)kS * kHID;   // 1024*1024
  float* Vm = Km + (size_t)kS * kHID;   // 1024*1024
  float* PQ = Vm + (size_t)kS * kHID;   // 256*1024
  float* PK = PQ + (size_t)kNS * kP * kHID;
  float* PV = PK + (size_t)kNS * kP * kHID;
  float* un = PV + (size_t)kNS * kP * kHID;  // 16*1024

  // 1) QKV projections (WMMA f32 GEMMs)
  qkv_kernel<<<dim3(kS / 16, 3 * kHID / 128), 256, 0, stream>>>(
      hs, Wqkv, qb, vb, Qm, Km, Vm);
  qkv_kernel<<<dim3(kNS * kP / 16, 3 * kHID / 128), 256, 0, stream>>>(
      pseq, Wqkv, qb, vb, PQ, PK, PV);

  // 2) key-unary bias
  unary_kernel<<<kNH * kS / 8, 256, 0, stream>>>(hs, Wun, un);

  // 3) fused flash attention, main path
  attn_main_kernel<<<dim3(kS / 16, kNH), 256, 0, stream>>>(
      Qm, Km, Vm, rel, Wrel, un, amask, out);

  // 4) fused flash attention, two-stream predict path (merged keys)
  attn_pred_kernel<<<dim3(kP / 16, kNH, kNS), 256, 0, stream>>>(
      PQ, PK, PV, Km, Vm, prel, Wrel, pmask, out + (size_t)kS * kHID);
}